// StableLapPENodeEncoder_26182120636871
// MI455X (gfx1250) — compile-verified
//
#include <hip/hip_runtime.h>
#include <hip/hip_bf16.h>

typedef __attribute__((ext_vector_type(2))) float v2f;
typedef __attribute__((ext_vector_type(8))) float v8f;
typedef __attribute__((ext_vector_type(4))) unsigned int u32x4;
typedef __attribute__((ext_vector_type(8))) int i32x8;
typedef __attribute__((ext_vector_type(4))) int i32x4;

#define F_ 32
#define DPE_ 64
#define DIN_ 512
#define DH_ 128   // 2*DPE
#define DXO_ 512  // DEMB - DPE
#define DEMB_ 576

#if defined(__HIP_DEVICE_COMPILE__) && __has_builtin(__builtin_amdgcn_tensor_load_to_lds)
#define USE_TDM 1
#else
#define USE_TDM 0
#endif

// ---------------------------------------------------------------------------
// Kernel 0: zero agg[N*32] and cnt[N] (contiguous region of N*33 floats)
// ---------------------------------------------------------------------------
__global__ void lap_init_kernel(float* buf, int n_elems) {
    int i = blockIdx.x * blockDim.x + threadIdx.x;
    if (i < n_elems) buf[i] = 0.0f;
}

// ---------------------------------------------------------------------------
// Kernel 1: edge scatter. One wave per edge, lane = frequency (F == wave32).
// ---------------------------------------------------------------------------
__global__ __launch_bounds__(256) void lap_edge_kernel(
    const float* __restrict__ eigvecs, const long long* __restrict__ edge_index,
    float* __restrict__ agg, float* __restrict__ cnt, int nedges) {
    int lane  = threadIdx.x & 31;
    int wave  = blockIdx.x * (blockDim.x >> 5) + (threadIdx.x >> 5);
    int nwave = gridDim.x * (blockDim.x >> 5);
    for (int e = wave; e < nedges; e += nwave) {
        long long s = edge_index[e];                      // src
        long long d = edge_index[(long long)nedges + e];  // dst
        float a = eigvecs[(size_t)d * F_ + lane];
        float b = eigvecs[(size_t)s * F_ + lane];
        float df = a - b;
        atomicAdd(&agg[(size_t)d * F_ + lane], df * df);
        if (lane == 0) atomicAdd(&cnt[d], 1.0f);
    }
}

// ---------------------------------------------------------------------------
// Kernel 2: per-node frequency sums (reference NaN-mask rules folded in).
// ---------------------------------------------------------------------------
__global__ void lap_node_sums_kernel(
    const float* __restrict__ agg, const float* __restrict__ cnt,
    const float* __restrict__ eigvals,
    float* __restrict__ s0o, float* __restrict__ s1o, float* __restrict__ nvo,
    int nnodes) {
    int n = blockIdx.x * blockDim.x + threadIdx.x;
    if (n >= nnodes) return;
    float c   = cnt[n];
    float inv = (c > 0.0f) ? (1.0f / fmaxf(c, 1.0f)) : 0.0f;
    float s0 = 0.0f, s1 = 0.0f, nv = 0.0f;
    #pragma unroll
    for (int f = 0; f < F_; ++f) {
        float p  = agg[(size_t)n * F_ + f] * inv;
        float ev = eigvals[(size_t)n * F_ + f];
        bool valid = !(p != p);
        if (valid) {
            s0 += p;
            s1 += (ev != ev) ? 0.0f : ev;
            nv += 1.0f;
        }
    }
    s0o[n] = s0; s1o[n] = s1; nvo[n] = nv;
}

// ---------------------------------------------------------------------------
// Kernel 3: post-MLP via f32 WMMA. One wave per 16-node group.
// ---------------------------------------------------------------------------
#define MLP_WAVES 4
__global__ __launch_bounds__(128) void lap_mlp_kernel(
    const float* __restrict__ s0a, const float* __restrict__ s1a,
    const float* __restrict__ nva,
    const float* __restrict__ WA, const float* __restrict__ bA,
    const float* __restrict__ W1, const float* __restrict__ b1,
    const float* __restrict__ W2, const float* __restrict__ b2,
    float* __restrict__ out, int nnodes) {
    __shared__ float pe0[MLP_WAVES][16 * DPE_];
    __shared__ float h1 [MLP_WAVES][16 * DH_];
    const int w    = threadIdx.x >> 5;
    const int lane = threadIdx.x & 31;
    const int l16  = lane & 15;
    const int koff = (lane < 16) ? 0 : 2;
    const int mbase = (lane < 16) ? 0 : 8;
    const int ngroups = nnodes >> 4;
    const int group = blockIdx.x * MLP_WAVES + w;
    const int node0 = group * 16;

    if (group < ngroups) {
        for (int idx = lane; idx < 16 * DPE_; idx += 32) {
            int i = idx >> 6, j = idx & 63;
            int n = node0 + i;
            pe0[w][idx] = s0a[n] * WA[j] + s1a[n] * WA[DPE_ + j] + nva[n] * bA[j];
        }
    }
    __syncthreads();

    if (group < ngroups) {
        #pragma unroll
        for (int nt = 0; nt < 8; ++nt) {
            const int col = nt * 16 + l16;
            v8f acc;
            float bias = b1[col];
            #pragma unroll
            for (int r = 0; r < 8; ++r) acc[r] = bias;
            #pragma unroll
            for (int k = 0; k < DPE_; k += 4) {
                v2f a, b;
                a.x = pe0[w][l16 * DPE_ + k + koff];
                a.y = pe0[w][l16 * DPE_ + k + koff + 1];
                b.x = W1[(k + koff) * DH_ + col];
                b.y = W1[(k + koff + 1) * DH_ + col];
                acc = __builtin_amdgcn_wmma_f32_16x16x4_f32(
                          false, a, false, b, (short)0, acc, false, false);
            }
            #pragma unroll
            for (int r = 0; r < 8; ++r)
                h1[w][(mbase + r) * DH_ + col] = fmaxf(acc[r], 0.0f);
        }
    }
    __syncthreads();

    if (group < ngroups) {
        #pragma unroll
        for (int nt = 0; nt < 4; ++nt) {
            const int col = nt * 16 + l16;
            v8f acc;
            float bias = b2[col];
            #pragma unroll
            for (int r = 0; r < 8; ++r) acc[r] = bias;
            #pragma unroll
            for (int k = 0; k < DH_; k += 4) {
                v2f a, b;
                a.x = h1[w][l16 * DH_ + k + koff];
                a.y = h1[w][l16 * DH_ + k + koff + 1];
                b.x = W2[(k + koff) * DPE_ + col];
                b.y = W2[(k + koff + 1) * DPE_ + col];
                acc = __builtin_amdgcn_wmma_f32_16x16x4_f32(
                          false, a, false, b, (short)0, acc, false, false);
            }
            #pragma unroll
            for (int r = 0; r < 8; ++r)
                out[(size_t)(node0 + mbase + r) * DEMB_ + DXO_ + col] =
                    fmaxf(acc[r], 0.0f);
        }
    }
}

// ---------------------------------------------------------------------------
// TDM: load one 32(K-rows) x 128(cols) f32 panel of Wx into LDS.
// D# per ISA ch.8: group0 = {count=1 | lds_addr | global_addr | type=2},
// group1 = {data_size=4B, tensor_dim0/1=512, tile_dim0=128, tile_dim1=32,
//           tensor_dim0_stride=512}. Groups 2/3 zero (2D tensor).
// ---------------------------------------------------------------------------
#if USE_TDM
__device__ __forceinline__ void tdm_load_panel(unsigned lds_off, const float* gsrc) {
    unsigned long long ga = (unsigned long long)(size_t)gsrc;
    u32x4 g0;
    g0[0] = 1u;                                             // count=1, user D#
    g0[1] = lds_off;                                        // lds_addr (bytes)
    g0[2] = (unsigned)(ga & 0xFFFFFFFFu);                   // global_addr[31:0]
    g0[3] = (unsigned)((ga >> 32) & 0x1FFFFFFu) | (2u << 30); // ga[56:32] | type=2
    i32x8 g1;
    g1[0] = (int)(2u << 16);       // workgroup_mask=0, data_size=2 (4 bytes)
    g1[1] = (int)(512u << 16);     // tensor_dim0[15:0]=512 in dw1[31:16]
    g1[2] = (int)(512u << 16);     // tensor_dim0 hi=0 | tensor_dim1[15:0]=512
    g1[3] = (int)(128u << 16);     // tensor_dim1 hi=0 | tile_dim0=128
    g1[4] = 32;                    // tile_dim1=32, tile_dim2=0
    g1[5] = 512;                   // tensor_dim0_stride = 512 elements
    g1[6] = 0;
    g1[7] = 0;
    i32x4 z4 = {0, 0, 0, 0};
#if defined(__clang_major__) && (__clang_major__ >= 23)
    i32x8 z8 = {0, 0, 0, 0, 0, 0, 0, 0};
    __builtin_amdgcn_tensor_load_to_lds(g0, g1, z4, z4, z8, 0);
#else
    __builtin_amdgcn_tensor_load_to_lds(g0, g1, z4, z4, 0);
#endif
}
#endif

// ---------------------------------------------------------------------------
// Kernel 4: h = x @ Wx + bx via f32 WMMA.
//   Block = 128 rows x 128 cols, 8 waves, each wave = 64 rows x 32 cols
//   (4 M-tiles x 2 N-tiles -> B fragment reused 4x, A loads are b64).
//   Wx K-panels staged in LDS via TDM (double buffered, s_wait_tensorcnt).
// ---------------------------------------------------------------------------
__global__ __launch_bounds__(256) void lap_gemm_x_kernel(
    const float* __restrict__ x, const float* __restrict__ Wx,
    const float* __restrict__ bx, float* __restrict__ out, int nnodes) {
    __shared__ float bpanel[2][32 * 128];                 // 32 KB
    const int tid  = threadIdx.x;
    const int w    = tid >> 5;
    const int lane = tid & 31;
    const int l16  = lane & 15;
    const int koff  = (lane < 16) ? 0 : 2;
    const int mbase = (lane < 16) ? 0 : 8;
    const int mb = blockIdx.x >> 2, nb = blockIdx.x & 3;
    const int row0 = mb * 128 + (w >> 2) * 64;            // wave's first row
    const int cloc = (w & 3) * 32 + l16;                  // col within 128 panel
    const int col0 = nb * 128;

    // accumulators: acc[nt][mt], C init = bias broadcast over M
    v8f acc[2][4];
    #pragma unroll
    for (int nt = 0; nt < 2; ++nt) {
        float bias = bx[col0 + (w & 3) * 32 + nt * 16 + l16];
        #pragma unroll
        for (int mt = 0; mt < 4; ++mt)
            #pragma unroll
            for (int r = 0; r < 8; ++r) acc[nt][mt][r] = bias;
    }

    // A row pointers (clamped for the partial last M-chunk; stores are guarded)
    const float* xr[4];
    #pragma unroll
    for (int mt = 0; mt < 4; ++mt) {
        int r = row0 + mt * 16 + l16;
        if (r > nnodes - 1) r = nnodes - 1;
        xr[mt] = x + (size_t)r * DIN_;
    }

#if USE_TDM
    unsigned ldsb = (unsigned)(size_t)(&bpanel[0][0]);
    if (w == 0) {
        tdm_load_panel(ldsb, Wx + col0);                  // chunk 0 -> buf 0
        __builtin_amdgcn_s_wait_tensorcnt(0);
    }
    __syncthreads();
#endif

    for (int kc = 0; kc < 16; ++kc) {                     // K chunks of 32
#if USE_TDM
        const float* bp = &bpanel[kc & 1][0];
        if (w == 0 && kc + 1 < 16)                        // overlap next panel
            tdm_load_panel(ldsb + (unsigned)(((kc + 1) & 1) * 32 * 128 * 4),
                           Wx + (size_t)((kc + 1) * 32) * DXO_ + col0);
#else
        const float* bp = &bpanel[0][0];
        __syncthreads();
        {   // cooperative fallback: 256 threads load 32x128 floats as float4
            const float4* src = (const float4*)(Wx + (size_t)(kc * 32) * DXO_ + col0);
            float4* dst = (float4*)&bpanel[0][0];
            for (int i = tid; i < 32 * 32; i += 256) {
                int rr = i >> 5, cc = i & 31;
                dst[rr * 32 + cc] = src[rr * 128 + cc];
            }
        }
        __syncthreads();
#endif
        const int k0 = kc * 32;
        #pragma unroll
        for (int mt = 0; mt < 4; ++mt)
            __builtin_prefetch(xr[mt] + k0 + 32, 0, 1);   // global_prefetch_b8

        #pragma unroll
        for (int kk = 0; kk < 32; kk += 4) {
            v2f a[4];
            #pragma unroll
            for (int mt = 0; mt < 4; ++mt)
                a[mt] = *(const v2f*)(xr[mt] + k0 + kk + koff);
            #pragma unroll
            for (int nt = 0; nt < 2; ++nt) {
                v2f b;
                b.x = bp[(kk + koff) * 128 + cloc + nt * 16];
                b.y = bp[(kk + koff + 1) * 128 + cloc + nt * 16];
                #pragma unroll
                for (int mt = 0; mt < 4; ++mt)
                    acc[nt][mt] = __builtin_amdgcn_wmma_f32_16x16x4_f32(
                        false, a[mt], false, b, (short)0, acc[nt][mt], false, false);
            }
        }
#if USE_TDM
        if (w == 0 && kc + 1 < 16) __builtin_amdgcn_s_wait_tensorcnt(0);
        __syncthreads();
#endif
    }

    #pragma unroll
    for (int nt = 0; nt < 2; ++nt)
        #pragma unroll
        for (int mt = 0; mt < 4; ++mt)
            #pragma unroll
            for (int r = 0; r < 8; ++r) {
                int row = row0 + mt * 16 + mbase + r;
                if (row < nnodes)
                    out[(size_t)row * DEMB_ + col0 + (w & 3) * 32 + nt * 16 + l16] =
                        acc[nt][mt][r];
            }
}

// ---------------------------------------------------------------------------
// Launch
// ---------------------------------------------------------------------------
extern "C" void kernel_launch(void* const* d_in, const int* in_sizes, int n_in,
                              void* d_out, int out_size, void* d_ws, size_t ws_size,
                              hipStream_t stream) {
    const float*     EigVecs = (const float*)d_in[0];
    const float*     EigVals = (const float*)d_in[1];
    const float*     x       = (const float*)d_in[2];
    const long long* eidx    = (const long long*)d_in[3];  // int64 per reference
    const float*     W_A     = (const float*)d_in[4];
    const float*     b_A     = (const float*)d_in[5];
    const float*     W1      = (const float*)d_in[6];
    const float*     b1      = (const float*)d_in[7];
    const float*     W2      = (const float*)d_in[8];
    const float*     b2      = (const float*)d_in[9];
    const float*     Wx      = (const float*)d_in[10];
    const float*     bx      = (const float*)d_in[11];
    float*           out     = (float*)d_out;

    const int N = in_sizes[0] / F_;        // 50000
    const int E = in_sizes[3] / 2;         // 1.6M

    float* agg = (float*)d_ws;             // N*32
    float* cnt = agg + (size_t)N * F_;     // N
    float* s0  = cnt + N;                  // N
    float* s1  = s0 + N;                   // N
    float* nv  = s1 + N;                   // N

    // 0) zero accumulators (every call: graph-replay safe)
    {
        int elems = N * (F_ + 1);
        lap_init_kernel<<<(elems + 255) / 256, 256, 0, stream>>>(agg, elems);
    }
    // 1) edge scatter
    lap_edge_kernel<<<2048, 256, 0, stream>>>(EigVecs, eidx, agg, cnt, E);
    // 2) per-node frequency sums
    lap_node_sums_kernel<<<(N + 255) / 256, 256, 0, stream>>>(
        agg, cnt, EigVals, s0, s1, nv, N);
    // 3) post-MLP -> out[:, 512:576]
    {
        int ngroups = N >> 4;
        int blocks  = (ngroups + MLP_WAVES - 1) / MLP_WAVES;
        lap_mlp_kernel<<<blocks, 32 * MLP_WAVES, 0, stream>>>(
            s0, s1, nv, W_A, b_A, W1, b1, W2, b2, out, N);
    }
    // 4) x @ Wx + bx -> out[:, 0:512]
    {
        int mblocks = (N + 127) / 128;     // 391
        lap_gemm_x_kernel<<<mblocks * 4, 256, 0, stream>>>(x, Wx, bx, out, N);
    }
}